// MultiHeadAttentionBlock_37890201485779
// MI455X (gfx1250) — compile-verified
//
#include <hip/hip_runtime.h>
#include <hip/hip_bf16.h>

typedef __attribute__((ext_vector_type(2))) float v2f;
typedef __attribute__((ext_vector_type(4))) float v4f;
typedef __attribute__((ext_vector_type(8))) float v8f;
typedef __attribute__((ext_vector_type(4))) int   v4i;

#define H 16
#define D 64
#define T 512
#define S 512
#define BQ 2
#define MD 1024

// ---- CDNA5 async global->LDS path (guarded: falls back to VGPR staging) ----
#if defined(__has_builtin)
#if __has_builtin(__builtin_amdgcn_global_load_async_to_lds_b128) && \
    __has_builtin(__builtin_amdgcn_s_wait_asynccnt)
#define USE_ASYNC_LDS 1
#endif
#endif
#ifndef USE_ASYNC_LDS
#define USE_ASYNC_LDS 0
#endif

// Copy 16 contiguous bytes global -> LDS (no transpose).
static __device__ __forceinline__ void cp16(float* l, const float* g) {
#if USE_ASYNC_LDS
    __builtin_amdgcn_global_load_async_to_lds_b128(
        (__attribute__((address_space(1))) v4i*)(g),
        (__attribute__((address_space(3))) v4i*)(l), 0, 0);
#else
    *reinterpret_cast<v4f*>(l) = *reinterpret_cast<const v4f*>(g);
#endif
}

static __device__ __forceinline__ void async_fence() {
#if USE_ASYNC_LDS
    __builtin_amdgcn_s_wait_asynccnt(0);
#endif
}

static __device__ __forceinline__ v8f wmma4(v2f a, v2f b, v8f c) {
    // D = A(16x4) * B(4x16) + C, fp32, full precision
    return __builtin_amdgcn_wmma_f32_16x16x4_f32(false, a, false, b, (short)0, c, false, false);
}

// ---------------------------------------------------------------------------
// Generic fp32 WMMA GEMM: C = A(MxK) * B(KxN) (B optionally transposed).
// Block = 128 threads (4 waves), block tile = 64x64, BK = 32.
// cmode: 0 = C[batch*cStride + m*ldc + n]
//        1 = head-split projection out: row m=(b,t), col n=(h,d)
//        2 = vals out for batched att@vh: batch=(b,h), row=t, col=d
// ---------------------------------------------------------------------------
__global__ __launch_bounds__(128) void gemm_f32_wmma(
    const float* __restrict__ A, int lda, long long aStride,
    const float* __restrict__ Bm, int ldb, long long bStride, int transB,
    float* __restrict__ C, int ldc, long long cStride, int cmode,
    int M, int N, int K)
{
    __shared__ float as[32 * 65];  // k-major: as[k][m], pad 1
    __shared__ float bs[32 * 66];  // k-major: bs[k][n], pad 2

    const int tid  = threadIdx.x;
    const int lane = tid & 31;
    const int w    = tid >> 5;
    const int lr   = lane & 15;
    const int kh2  = (lane >> 4) << 1;   // A/B K sub-offset per ISA layout
    const int mhi  = (lane >> 4) << 3;   // C/D row offset for upper half-wave

    const int batch = blockIdx.z;
    const int m0 = blockIdx.y * 64;
    const int n0 = blockIdx.x * 64;
    const float* Ab = A  + (long long)batch * aStride;
    const float* Bb = Bm + (long long)batch * bStride;

    v8f acc[4];
#pragma unroll
    for (int i = 0; i < 4; ++i) acc[i] = (v8f){0.f,0.f,0.f,0.f,0.f,0.f,0.f,0.f};

    const int mw = w * 16;
    const int nChunks = K >> 5;
    for (int c = 0; c < nChunks; ++c) {
        const int k0 = c << 5;
        // ---- stage B: bs[k][n] ----
        if (!transB) {
            // natural layout: 16B chunks contiguous on both sides -> async DMA
#pragma unroll
            for (int i = 0; i < 4; ++i) {
                int v = tid + i * 128;        // [0,512)
                int k = v >> 4;
                int n = (v & 15) << 2;
                cp16(&bs[k * 66 + n], &Bb[(long long)(k0 + k) * ldb + n0 + n]);
            }
        } else {  // B[k][n] = Bm[n*ldb + k] : transpose via VGPRs
#pragma unroll
            for (int i = 0; i < 4; ++i) {
                int v = tid + i * 128;
                int n = v >> 3;
                int k = (v & 7) << 2;
                v4f t4 = *reinterpret_cast<const v4f*>(&Bb[(long long)(n0 + n) * ldb + k0 + k]);
                bs[(k + 0) * 66 + n] = t4.x;
                bs[(k + 1) * 66 + n] = t4.y;
                bs[(k + 2) * 66 + n] = t4.z;
                bs[(k + 3) * 66 + n] = t4.w;
            }
        }
        // ---- stage A: 64 m x 32 k, transposed into as[k][m] ----
#pragma unroll
        for (int i = 0; i < 4; ++i) {
            int v = tid + i * 128;            // [0,512)
            int m = v >> 3;
            int k = (v & 7) << 2;
            v4f t4 = *reinterpret_cast<const v4f*>(&Ab[(long long)(m0 + m) * lda + k0 + k]);
            as[(k + 0) * 65 + m] = t4.x;
            as[(k + 1) * 65 + m] = t4.y;
            as[(k + 2) * 65 + m] = t4.z;
            as[(k + 3) * 65 + m] = t4.w;
        }
        if (!transB) async_fence();
        __syncthreads();
        // ---- WMMA compute ----
#pragma unroll
        for (int kk = 0; kk < 32; kk += 4) {
            const int kA = kk + kh2;
            v2f a;
            a.x = as[(kA + 0) * 65 + mw + lr];
            a.y = as[(kA + 1) * 65 + mw + lr];
#pragma unroll
            for (int nt = 0; nt < 4; ++nt) {
                v2f b;
                b.x = bs[(kA + 0) * 66 + nt * 16 + lr];
                b.y = bs[(kA + 1) * 66 + nt * 16 + lr];
                acc[nt] = wmma4(a, b, acc[nt]);
            }
        }
        __syncthreads();
    }

    // ---- store per cmode ----
#pragma unroll
    for (int r = 0; r < 8; ++r) {
        const int m = m0 + mw + r + mhi;
#pragma unroll
        for (int nt = 0; nt < 4; ++nt) {
            const int n = n0 + nt * 16 + lr;
            const float val = acc[nt][r];
            size_t idx;
            if (cmode == 0) {
                idx = (size_t)batch * (size_t)cStride + (size_t)m * ldc + n;
            } else if (cmode == 1) {
                // row m = b*T + t ; col n = h*64 + d -> qh[((b*H+h)*T+t)*D+d]
                idx = ((((size_t)(m >> 9) * H + (n >> 6)) * T + (m & 511)) << 6) + (n & 63);
            } else {
                // batch = b*H + h ; m = t ; n = d -> vals[(b*T+t)*MD + h*64 + n]
                idx = (((size_t)(batch >> 4) * T + m) << 10) + (size_t)((batch & 15) << 6) + n;
            }
            C[idx] = val;
        }
    }
}

// ---------------------------------------------------------------------------
// Extra logit terms per fixed (b,t): M = 16 heads.
// logits[b,h,t,s] = ( logits + qh[b,h,t,:].ak[b,t,s,:] + kh[b,h,t,:].aq[b,s,t,:] )
//                   * exp(min(ls[h],log100)) + bias[b,t,s,h], then * 1/8.
// ak/aq/bias are single-use streams -> non-temporal loads.
// Block = 128 threads (4 waves), covers 64 s columns. grid = (S/64, B*T)
// ---------------------------------------------------------------------------
__global__ __launch_bounds__(128) void extra_logits_kernel(
    const float* __restrict__ qh, const float* __restrict__ kh,
    const float* __restrict__ ak, const float* __restrict__ aq,
    const float* __restrict__ bias, const float* __restrict__ lsc,
    float* __restrict__ logits)
{
    __shared__ float as1[64 * 17];  // qh^T : [d][h]
    __shared__ float as2[64 * 17];  // kh^T : [d][h]
    __shared__ float bs1[64 * 66];  // ak^T : [d][s]
    __shared__ float bs2[64 * 66];  // aq^T : [d][j]

    const int tid  = threadIdx.x;
    const int lane = tid & 31;
    const int w    = tid >> 5;
    const int lr   = lane & 15;
    const int kh2  = (lane >> 4) << 1;
    const int mhi  = (lane >> 4) << 3;

    const int bt = blockIdx.y;
    const int b  = bt >> 9;
    const int t  = bt & 511;
    const int s0 = blockIdx.x * 64;

    // stage A1 (qh) and A2 (kh): 16h x 64d each
#pragma unroll
    for (int i = 0; i < 2; ++i) {
        int v = tid + i * 128;             // [0,256)
        int h = v >> 4;
        int d = (v & 15) << 2;
        size_t qi = ((((size_t)(b * H + h)) * T + t) << 6) + d;
        v4f t1 = *reinterpret_cast<const v4f*>(&qh[qi]);
        v4f t2 = *reinterpret_cast<const v4f*>(&kh[qi]);
        as1[(d + 0) * 17 + h] = t1.x; as1[(d + 1) * 17 + h] = t1.y;
        as1[(d + 2) * 17 + h] = t1.z; as1[(d + 3) * 17 + h] = t1.w;
        as2[(d + 0) * 17 + h] = t2.x; as2[(d + 1) * 17 + h] = t2.y;
        as2[(d + 2) * 17 + h] = t2.z; as2[(d + 3) * 17 + h] = t2.w;
    }
    // stage B1 = ak[b,t,s0+si,:]^T  and  B2 = aq[b,s0+ji,t,:]^T  (non-temporal)
#pragma unroll
    for (int i = 0; i < 8; ++i) {
        int v  = tid + i * 128;            // [0,1024)
        int si = v >> 4;
        int d  = (v & 15) << 2;
        size_t ai = ((((size_t)b * T + t) * S + (s0 + si)) << 6) + d;
        size_t gi = ((((size_t)b * T + (s0 + si)) * S + t) << 6) + d;
        v4f t1 = __builtin_nontemporal_load(reinterpret_cast<const v4f*>(&ak[ai]));
        v4f t2 = __builtin_nontemporal_load(reinterpret_cast<const v4f*>(&aq[gi]));
        bs1[(d + 0) * 66 + si] = t1.x; bs1[(d + 1) * 66 + si] = t1.y;
        bs1[(d + 2) * 66 + si] = t1.z; bs1[(d + 3) * 66 + si] = t1.w;
        bs2[(d + 0) * 66 + si] = t2.x; bs2[(d + 1) * 66 + si] = t2.y;
        bs2[(d + 2) * 66 + si] = t2.z; bs2[(d + 3) * 66 + si] = t2.w;
    }
    __syncthreads();

    v8f acc = (v8f){0.f,0.f,0.f,0.f,0.f,0.f,0.f,0.f};
    const int sCol = w * 16;
#pragma unroll
    for (int kk = 0; kk < 64; kk += 4) {
        const int kA = kk + kh2;
        v2f a1, b1, a2, b2;
        a1.x = as1[(kA + 0) * 17 + lr];        a1.y = as1[(kA + 1) * 17 + lr];
        b1.x = bs1[(kA + 0) * 66 + sCol + lr]; b1.y = bs1[(kA + 1) * 66 + sCol + lr];
        acc = wmma4(a1, b1, acc);
        a2.x = as2[(kA + 0) * 17 + lr];        a2.y = as2[(kA + 1) * 17 + lr];
        b2.x = bs2[(kA + 0) * 66 + sCol + lr]; b2.y = bs2[(kA + 1) * 66 + sCol + lr];
        acc = wmma4(a2, b2, acc);
    }

#pragma unroll
    for (int r = 0; r < 8; ++r) {
        const int h = r + mhi;
        const int s = s0 + sCol + lr;
        const size_t li = (((size_t)(b * H + h) * T + t) << 9) + s;
        const float sc = __expf(fminf(lsc[h], 4.6051702f));  // min(ls, log 100)
        float val = (logits[li] + acc[r]) * sc;
        val += __builtin_nontemporal_load(&bias[((((size_t)b * T + t) * S + s) << 4) + h]);
        logits[li] = val * 0.125f;                            // / sqrt(64)
    }
}

// ---------------------------------------------------------------------------
// Row softmax over S=512. 1 wave per row, 8 rows per block.
// ---------------------------------------------------------------------------
__global__ __launch_bounds__(256) void softmax_kernel(float* __restrict__ att)
{
    const int lane = threadIdx.x & 31;
    const int row  = blockIdx.x * 8 + (threadIdx.x >> 5);
    const size_t base = (size_t)row * S;
    float v[16];
    float mx = -INFINITY;
#pragma unroll
    for (int i = 0; i < 16; ++i) { v[i] = att[base + i * 32 + lane]; mx = fmaxf(mx, v[i]); }
#pragma unroll
    for (int off = 16; off > 0; off >>= 1) mx = fmaxf(mx, __shfl_xor(mx, off, 32));
    float sum = 0.f;
#pragma unroll
    for (int i = 0; i < 16; ++i) { v[i] = __expf(v[i] - mx); sum += v[i]; }
#pragma unroll
    for (int off = 16; off > 0; off >>= 1) sum += __shfl_xor(sum, off, 32);
    const float inv = 1.0f / sum;
#pragma unroll
    for (int i = 0; i < 16; ++i) att[base + i * 32 + lane] = v[i] * inv;
}

// ---------------------------------------------------------------------------
// vals[b,t,h,d] += sum_s att[b,h,t,s] * av[b,t,s,d]. Per fixed (b,t): M=16 heads,
// K = S = 512, N = 64. Block = 128 threads (4 waves, 16 d-cols each).
// av tile is natural-layout -> async global->LDS DMA.
// ---------------------------------------------------------------------------
__global__ __launch_bounds__(128) void av_values_kernel(
    const float* __restrict__ att, const float* __restrict__ av,
    float* __restrict__ vals)
{
    __shared__ float ask[64 * 17];  // att^T chunk: [s][h]
    __shared__ float bsv[64 * 66];  // av chunk:    [s][d]

    const int tid  = threadIdx.x;
    const int lane = tid & 31;
    const int w    = tid >> 5;
    const int lr   = lane & 15;
    const int kh2  = (lane >> 4) << 1;
    const int mhi  = (lane >> 4) << 3;

    const int bt = blockIdx.x;
    const int b  = bt >> 9;
    const int t  = bt & 511;
    const int dCol = w * 16;

    v8f acc = (v8f){0.f,0.f,0.f,0.f,0.f,0.f,0.f,0.f};

    for (int k0 = 0; k0 < S; k0 += 64) {
        // stage av 64s x 64d (natural layout -> async copy)
#pragma unroll
        for (int i = 0; i < 8; ++i) {
            int v  = tid + i * 128;        // [0,1024)
            int si = v >> 4;
            int d  = (v & 15) << 2;
            size_t vidx = ((((size_t)b * T + t) * S + k0 + si) << 6) + d;
            cp16(&bsv[si * 66 + d], &av[vidx]);
        }
        // stage att 16h x 64s (transposed via VGPRs)
#pragma unroll
        for (int i = 0; i < 2; ++i) {
            int v = tid + i * 128;         // [0,256)
            int h = v >> 4;
            int sg = (v & 15) << 2;
            size_t aidx = (((size_t)(b * H + h) * T + t) << 9) + k0 + sg;
            v4f t4 = *reinterpret_cast<const v4f*>(&att[aidx]);
            ask[(sg + 0) * 17 + h] = t4.x; ask[(sg + 1) * 17 + h] = t4.y;
            ask[(sg + 2) * 17 + h] = t4.z; ask[(sg + 3) * 17 + h] = t4.w;
        }
        async_fence();
        __syncthreads();
#pragma unroll
        for (int kk = 0; kk < 64; kk += 4) {
            const int kA = kk + kh2;
            v2f a, bv;
            a.x  = ask[(kA + 0) * 17 + lr];        a.y  = ask[(kA + 1) * 17 + lr];
            bv.x = bsv[(kA + 0) * 66 + dCol + lr]; bv.y = bsv[(kA + 1) * 66 + dCol + lr];
            acc = wmma4(a, bv, acc);
        }
        __syncthreads();
    }

#pragma unroll
    for (int r = 0; r < 8; ++r) {
        const int h = r + mhi;
        const size_t vi = (((size_t)(b * T + t)) << 10) + (h << 6) + dCol + lr;
        vals[vi] += acc[r];
    }
}

// ---------------------------------------------------------------------------
extern "C" void kernel_launch(void* const* d_in, const int* in_sizes, int n_in,
                              void* d_out, int out_size, void* d_ws, size_t ws_size,
                              hipStream_t stream) {
    (void)in_sizes; (void)n_in; (void)out_size; (void)ws_size;
    const float* q    = (const float*)d_in[0];
    const float* k    = (const float*)d_in[1];
    const float* v    = (const float*)d_in[2];
    const float* ak   = (const float*)d_in[3];
    const float* av   = (const float*)d_in[4];
    const float* aq   = (const float*)d_in[5];
    const float* bias = (const float*)d_in[6];
    const float* Wq   = (const float*)d_in[7];
    const float* Wk   = (const float*)d_in[8];
    const float* Wv   = (const float*)d_in[9];
    const float* Wo   = (const float*)d_in[10];
    const float* lsc  = (const float*)d_in[11];
    float* out = (float*)d_out;

    float* ws     = (float*)d_ws;
    float* qh     = ws;                 // 1M floats: (b,h,t,d)
    float* kh     = ws + 1048576;       // 1M
    float* vh     = ws + 2097152;       // 1M
    float* logits = ws + 3145728;       // 8M: (b,h,t,s) -> becomes att
    float* vals   = ws + 11534336;      // 1M: (b,t, h*64+d)

    dim3 blk(128);
    // Projections with head-split epilogue: M=N=K=1024
    gemm_f32_wmma<<<dim3(16,16,1), blk, 0, stream>>>(q, MD, 0, Wq, MD, 0, 0, qh, 0, 0, 1, MD, MD, MD);
    gemm_f32_wmma<<<dim3(16,16,1), blk, 0, stream>>>(k, MD, 0, Wk, MD, 0, 0, kh, 0, 0, 1, MD, MD, MD);
    gemm_f32_wmma<<<dim3(16,16,1), blk, 0, stream>>>(v, MD, 0, Wv, MD, 0, 0, vh, 0, 0, 1, MD, MD, MD);
    // logits = qh @ kh^T per (b,h): batched, transB, M=T, N=S, K=64
    gemm_f32_wmma<<<dim3(8,8,BQ*H), blk, 0, stream>>>(qh, D, (long long)T*D,
                                                      kh, D, (long long)S*D, 1,
                                                      logits, S, (long long)T*S, 0, T, S, D);
    // extra terms + scale + bias + 1/sqrt(d)
    extra_logits_kernel<<<dim3(S/64, BQ*T), blk, 0, stream>>>(qh, kh, ak, aq, bias, lsc, logits);
    // softmax over s
    softmax_kernel<<<dim3(BQ*H*T/8), dim3(256), 0, stream>>>(logits);
    // vals = att @ vh per (b,h): M=T, N=64, K=S ; head-split-vals epilogue
    gemm_f32_wmma<<<dim3(1,8,BQ*H), blk, 0, stream>>>(logits, S, (long long)T*S,
                                                      vh, D, (long long)S*D, 0,
                                                      vals, 0, 0, 2, T, D, S);
    // vals += att . av  (per (b,t), M=16 heads)
    av_values_kernel<<<dim3(BQ*T), blk, 0, stream>>>(logits, av, vals);
    // out = vals @ Wo
    gemm_f32_wmma<<<dim3(16,16,1), blk, 0, stream>>>(vals, MD, 0, Wo, MD, 0, 0, out, MD, 0, 0, MD, MD, MD);
}